// gcn_lstm_cell_32049045962800
// MI455X (gfx1250) — compile-verified
//
#include <hip/hip_runtime.h>
#include <hip/hip_bf16.h>

// ---------------------------------------------------------------------------
// GCN-LSTM cell for MI455X (gfx1250, wave32).
//   Phase 1: zero agg/deg
//   Phase 2: degree scatter (edges + self loops), then dinv = rsqrt(deg)
//   Phase 3: xw = [x|h] @ [Wi|Wf|Wo|Wg]  via V_WMMA_F32_16X16X4_F32 (native f32 WMMA)
//   Phase 4: agg[dst] += dinv[src]*dinv[dst] * xw[src]  (wave per edge, f32 atomics @ L2)
//   Phase 5: gates + LSTM update -> d_out = h_new ++ c_new
// ---------------------------------------------------------------------------

typedef float v2f __attribute__((ext_vector_type(2)));
typedef float v8f __attribute__((ext_vector_type(8)));

#define N_FEAT   128          // n_in == n_out
#define K_DIM    256          // d_comb
#define N_OUT4   512          // 4 gates * 128

// ---------------------------------------------------------------- zero fill
__global__ void zero_f4_kernel(float4* __restrict__ p, long n4) {
    long i = (long)blockIdx.x * blockDim.x + threadIdx.x;
    if (i < n4) p[i] = make_float4(0.f, 0.f, 0.f, 0.f);
}

// ---------------------------------------------------------------- degree
__global__ void degree_kernel(const int* __restrict__ dst, float* __restrict__ deg,
                              int nEdges, int nNodes) {
    int i = blockIdx.x * blockDim.x + threadIdx.x;
    int total = nEdges + nNodes;
    if (i >= total) return;
    int d = (i < nEdges) ? dst[i] : (i - nEdges);   // self loops appended
    unsafeAtomicAdd(&deg[d], 1.0f);
}

__global__ void dinv_kernel(float* __restrict__ deg, int nNodes) {
    int i = blockIdx.x * blockDim.x + threadIdx.x;
    if (i >= nNodes) return;
    float d = deg[i];
    deg[i] = (d > 0.f) ? rsqrtf(d) : 0.f;           // in-place: deg -> dinv
}

// ---------------------------------------------------------------- f32 WMMA GEMM
// xw[M=10000, 512] = concat(x,h)[M,256] @ concat(Wi,Wf,Wo,Wg)[256,512]
// grid = (M/16, 8), block = 128 (4 waves); each wave owns one 16x16 tile,
// wave w covers n-tile = blockIdx.y*4 + w.
__global__ void gemm_xw_wmma(const float* __restrict__ x, const float* __restrict__ h,
                             const float* __restrict__ Wi, const float* __restrict__ Wf,
                             const float* __restrict__ Wo, const float* __restrict__ Wg,
                             float* __restrict__ xw) {
    const int lane  = threadIdx.x & 31;
    const int wave  = threadIdx.x >> 5;
    const int mbase = blockIdx.x * 16;
    const int ntile = blockIdx.y * 4 + wave;        // 0..31
    const int nglob = ntile * 16;

    // select gate weight block (each gate spans 128 = 8 n-tiles)
    const int gate = nglob >> 7;
    const int ncol = nglob & 127;
    const float* __restrict__ W = (gate == 0) ? Wi : (gate == 1) ? Wf
                                 : (gate == 2) ? Wo : Wg;        // [256,128] row-major

    const int l16  = lane & 15;
    const int koff = (lane >> 4) * 2;               // lanes 0-15: K+0/1, lanes 16-31: K+2/3
    const int m    = mbase + l16;                   // A fragment row
    const int ncl  = ncol + l16;                    // B fragment col within gate

    v8f acc = {};
    #pragma unroll 8
    for (int k0 = 0; k0 < K_DIM; k0 += 4) {
        // A = concat(x,h): k-quad never straddles the 128 boundary
        const float* Arow = (k0 < N_FEAT) ? (x + (size_t)m * N_FEAT + k0)
                                          : (h + (size_t)m * N_FEAT + (k0 - N_FEAT));
        v2f a = *(const v2f*)(Arow + koff);         // A[m][k0+koff], A[m][k0+koff+1]
        v2f b;
        b.x = W[(size_t)(k0 + koff)     * N_FEAT + ncl];
        b.y = W[(size_t)(k0 + koff + 1) * N_FEAT + ncl];
        // D = A(16x4) * B(4x16) + C   -> v_wmma_f32_16x16x4_f32
        acc = __builtin_amdgcn_wmma_f32_16x16x4_f32(
                  /*neg_a=*/false, a, /*neg_b=*/false, b,
                  /*c_mod=*/(short)0, acc, /*reuse_a=*/false, /*reuse_b=*/false);
    }

    // C/D layout: VGPR r -> row M=r (lanes 0-15) / M=r+8 (lanes 16-31), N = lane%16
    const int mofs = (lane >> 4) * 8;
    #pragma unroll
    for (int r = 0; r < 8; ++r) {
        xw[(size_t)(mbase + r + mofs) * N_OUT4 + nglob + l16] = acc[r];
    }
}

// ---------------------------------------------------------------- edge scatter
// One wave per edge: lane j handles float4 chunks j, j+32, j+64, j+96 of the
// 512-float row. Loads are fully coalesced; adds are native f32 atomics at L2.
__global__ void scatter_kernel(const int* __restrict__ src, const int* __restrict__ dst,
                               const float* __restrict__ dinv,
                               const float* __restrict__ xw, float* __restrict__ agg,
                               int nEdges, int nNodes) {
    int gid  = blockIdx.x * blockDim.x + threadIdx.x;
    int e    = gid >> 5;
    int lane = gid & 31;
    int total = nEdges + nNodes;
    if (e >= total) return;

    int s, d;
    if (e < nEdges) { s = src[e]; d = dst[e]; }
    else            { s = d = e - nEdges; }         // self loop
    float norm = dinv[s] * dinv[d];

    const float4* __restrict__ row = (const float4*)(xw + (size_t)s * N_OUT4);
    float*        __restrict__ out = agg + (size_t)d * N_OUT4;

    #pragma unroll
    for (int j = 0; j < 4; ++j) {
        int c4 = lane + 32 * j;                     // 0..127 float4 chunks
        float4 v = row[c4];
        int fb = c4 * 4;
        unsafeAtomicAdd(&out[fb + 0], v.x * norm);
        unsafeAtomicAdd(&out[fb + 1], v.y * norm);
        unsafeAtomicAdd(&out[fb + 2], v.z * norm);
        unsafeAtomicAdd(&out[fb + 3], v.w * norm);
    }
}

// ---------------------------------------------------------------- gates + LSTM
__global__ void gates_kernel(const float* __restrict__ agg, const float* __restrict__ c,
                             const float* __restrict__ bi, const float* __restrict__ bf,
                             const float* __restrict__ bo, const float* __restrict__ bg,
                             float* __restrict__ out, int nNodes) {
    int idx = blockIdx.x * blockDim.x + threadIdx.x;    // n*128 + j
    if (idx >= nNodes * N_FEAT) return;
    int n = idx >> 7, j = idx & 127;
    const float* row = agg + (size_t)n * N_OUT4;

    float i_ = row[j            ] + bi[j];
    float f_ = row[N_FEAT   + j ] + bf[j];
    float o_ = row[2*N_FEAT + j ] + bo[j];
    float g_ = row[3*N_FEAT + j ] + bg[j];

    i_ = 1.f / (1.f + expf(-i_));
    f_ = 1.f / (1.f + expf(-f_));
    o_ = 1.f / (1.f + expf(-o_));
    g_ = tanhf(g_);

    float cn = f_ * c[idx] + i_ * g_;
    float hn = o_ * tanhf(cn);

    out[idx] = hn;                                  // h_new first
    out[(size_t)nNodes * N_FEAT + idx] = cn;        // then c_new
}

// ---------------------------------------------------------------------------
extern "C" void kernel_launch(void* const* d_in, const int* in_sizes, int n_in,
                              void* d_out, int out_size, void* d_ws, size_t ws_size,
                              hipStream_t stream) {
    const float* x  = (const float*)d_in[0];
    const float* h  = (const float*)d_in[1];
    const float* c  = (const float*)d_in[2];
    const float* Wi = (const float*)d_in[3];
    const float* bi = (const float*)d_in[4];
    const float* Wf = (const float*)d_in[5];
    const float* bf = (const float*)d_in[6];
    const float* Wo = (const float*)d_in[7];
    const float* bo = (const float*)d_in[8];
    const float* Wg = (const float*)d_in[9];
    const float* bg = (const float*)d_in[10];
    const int*   ei = (const int*)d_in[11];

    const int nNodes = in_sizes[0] / N_FEAT;        // 10000 (multiple of 16)
    const int nEdges = in_sizes[11] / 2;            // 640000
    const int* src = ei;
    const int* dst = ei + nEdges;

    // workspace layout: xw | agg | deg(->dinv)
    float* xw  = (float*)d_ws;
    float* agg = xw  + (size_t)nNodes * N_OUT4;
    float* deg = agg + (size_t)nNodes * N_OUT4;

    // zero agg + deg (contiguous, both multiples of 4 floats)
    {
        long n4 = ((long)nNodes * N_OUT4 + nNodes) / 4;
        int blk = 256;
        long grid = (n4 + blk - 1) / blk;
        zero_f4_kernel<<<(unsigned)grid, blk, 0, stream>>>((float4*)agg, n4);
    }
    // degree
    {
        int total = nEdges + nNodes;
        int blk = 256;
        degree_kernel<<<(total + blk - 1) / blk, blk, 0, stream>>>(dst, deg, nEdges, nNodes);
        dinv_kernel<<<(nNodes + blk - 1) / blk, blk, 0, stream>>>(deg, nNodes);
    }
    // GEMM: grid (M/16, 8), 4 waves per block each owning one 16x16 tile
    {
        dim3 grid(nNodes / 16, (N_OUT4 / 16) / 4);
        gemm_xw_wmma<<<grid, 128, 0, stream>>>(x, h, Wi, Wf, Wo, Wg, xw);
    }
    // edge scatter: one wave per (edge or self-loop)
    {
        long waves = (long)nEdges + nNodes;
        long threads = waves * 32;
        int blk = 256;
        long grid = (threads + blk - 1) / blk;
        scatter_kernel<<<(unsigned)grid, blk, 0, stream>>>(src, dst, deg, xw, agg,
                                                           nEdges, nNodes);
    }
    // gates
    {
        int total = nNodes * N_FEAT;
        int blk = 256;
        gates_kernel<<<(total + blk - 1) / blk, blk, 0, stream>>>(agg, c, bi, bf, bo, bg,
                                                                  (float*)d_out, nNodes);
    }
}